// Distral_80401787781327
// MI455X (gfx1250) — compile-verified
//
#include <hip/hip_runtime.h>

// CDNA5 / gfx1250: wave32, WMMA f32 16x16x4, branchless operand staging.
typedef float v2f __attribute__((ext_vector_type(2)));
typedef float v8f __attribute__((ext_vector_type(8)));

#define IN_DIM 3
#define HID 100
#define OUT_DIM 5
#define WAVES_PER_BLOCK 8

__global__ __launch_bounds__(256) void distral_mlp_wmma(
    const float* __restrict__ x,
    const float* __restrict__ W1,
    const float* __restrict__ b1,
    const float* __restrict__ W2,
    const float* __restrict__ b2,
    float* __restrict__ out,
    int n_heads)
{
    // 112 floats of h per wave (7 chunks of 16; only h[0..99] is ever read).
    __shared__ float hbuf[WAVES_PER_BLOCK][112];

    const int lane = threadIdx.x & 31;
    const int wave = threadIdx.x >> 5;
    const int head = blockIdx.x * WAVES_PER_BLOCK + wave;
    if (head >= n_heads) return;   // uniform per wave -> EXEC all-1s at WMMAs

    const float* W1_t = W1 + (size_t)head * (HID * IN_DIM);
    const float* b1_t = b1 + (size_t)head * HID;
    const float* W2_t = W2 + (size_t)head * (OUT_DIM * HID);
    const float* b2_t = b2 + (size_t)head * OUT_DIM;
    const float* x_t  = x  + (size_t)head * IN_DIM;
    float*       o_t  = out + (size_t)head * OUT_DIM;

    const int  L    = lane & 15;
    const bool hi   = lane >= 16;           // K-high half (K=2,3) for 32-bit A/B layout
    const int  off0 = hi ? 2 : 0;
    float* hw = hbuf[wave];

    // ---------------- Layer 1: h = relu(W1 @ x + b1) ----------------
    // Per-lane base pointers; each unrolled chunk uses pure immediate offsets.
    const float* a0p = W1_t + L * 3 + off0;          // K0 (lo) / K2 (hi)
    const float* a1p = a0p + (hi ? 0 : 1);           // K1 (lo) / dup K2 (hi, masked via B)
    const float* b1p = b1_t + (hi ? 8 : 0);          // bias rows r / 8+r

    // B (4x16): column 0 = x. Other columns replicate -> unread, harmless.
    // Column 0's K=3 slot MUST be 0 (pairs with a1p garbage on hi lanes).
    const float x0 = x_t[0], x1 = x_t[1], x2 = x_t[2];
    v2f Bx; Bx.x = hi ? x2 : x0; Bx.y = hi ? 0.f : x1;

    const bool is016 = (L == 0);                     // lanes 0 and 16
    float* hst = hw + (hi ? 8 : 0);                  // h store base for this lane
    const float inf = __builtin_inff();

#pragma unroll
    for (int c = 0; c < 7; ++c) {
        // A: rows c*16+L. Rows >= 100 (c==6) produce garbage h[100..111]: never read.
        const float a0 = a0p[c * 48];
        const float a1 = a1p[c * 48];
        v2f Av; Av.x = a0; Av.y = a1;

        // Bias pre-loaded into C (all lanes; unread rows/cols harmless).
        const float4 bl = *(const float4*)(b1p + c * 16);
        const float4 bh = *(const float4*)(b1p + c * 16 + 4);
        v8f acc;
        acc[0] = bl.x; acc[1] = bl.y; acc[2] = bl.z; acc[3] = bl.w;
        acc[4] = bh.x; acc[5] = bh.y; acc[6] = bh.z; acc[7] = bh.w;

        acc = __builtin_amdgcn_wmma_f32_16x16x4_f32(
                  false, Av, false, Bx, (short)0, acc, false, false);

        // ReLU = clamp to [0, inf): single v_med3_num_f32 per element.
#pragma unroll
        for (int r = 0; r < 8; ++r) acc[r] = __builtin_amdgcn_fmed3f(acc[r], 0.f, inf);

        if (is016) {
            float4 lo4, hi4;
            lo4.x = acc[0]; lo4.y = acc[1]; lo4.z = acc[2]; lo4.w = acc[3];
            hi4.x = acc[4]; hi4.y = acc[5]; hi4.z = acc[6]; hi4.w = acc[7];
            *(float4*)(hst + c * 16)     = lo4;
            *(float4*)(hst + c * 16 + 4) = hi4;
        }
    }
    // Cross-lane handoff (lane16 stores -> lane0 loads) through LDS: fence DS.
    asm volatile("s_wait_dscnt 0" ::: "memory");

    // ---------------- Layer 2: logits = W2 @ h + b2 ----------------
    // A: W2 rows, lanes with L>=5 clamp to row 0 (their D rows 5..15 unread).
    const int Lc = (L < OUT_DIM) ? L : 0;
    const float* w2p = W2_t + Lc * HID + off0;
    const float* hp  = hw + off0;                    // B: h chunk, all lanes

    v8f accA, accB;
#pragma unroll
    for (int m = 0; m < 8; ++m) { accA[m] = 0.f; accB[m] = 0.f; }
#pragma unroll
    for (int m = 0; m < OUT_DIM; ++m) accA[m] = b2_t[m];  // bias in C (all lanes)

#pragma unroll
    for (int kk = 0; kk < HID; kk += 4) {
        const float2 w  = *(const float2*)(w2p + kk);
        const float2 hv = *(const float2*)(hp + kk);
        v2f Av; Av.x = w.x;  Av.y = w.y;
        v2f Bv; Bv.x = hv.x; Bv.y = hv.y;
        if ((kk >> 2) & 1)
            accB = __builtin_amdgcn_wmma_f32_16x16x4_f32(
                       false, Av, false, Bv, (short)0, accB, false, false);
        else
            accA = __builtin_amdgcn_wmma_f32_16x16x4_f32(
                       false, Av, false, Bv, (short)0, accA, false, false);
    }

    // ---------------- Softmax (lane 0 holds logits[0..4]) ----------------
    if (lane == 0) {
        float lg[OUT_DIM];
#pragma unroll
        for (int m = 0; m < OUT_DIM; ++m) lg[m] = accA[m] + accB[m];

        float mx = lg[0];
#pragma unroll
        for (int m = 1; m < OUT_DIM; ++m) mx = __builtin_fmaxf(mx, lg[m]);

        float e[OUT_DIM], s = 0.f;
#pragma unroll
        for (int m = 0; m < OUT_DIM; ++m) { e[m] = __expf(lg[m] - mx); s += e[m]; }

        const float inv = 1.f / s;
#pragma unroll
        for (int m = 0; m < OUT_DIM; ++m) o_t[m] = e[m] * inv;
    }
}

extern "C" void kernel_launch(void* const* d_in, const int* in_sizes, int n_in,
                              void* d_out, int out_size, void* d_ws, size_t ws_size,
                              hipStream_t stream) {
    (void)n_in; (void)out_size; (void)d_ws; (void)ws_size;
    const float* x  = (const float*)d_in[0];
    const float* W1 = (const float*)d_in[1];
    const float* b1 = (const float*)d_in[2];
    const float* W2 = (const float*)d_in[3];
    const float* b2 = (const float*)d_in[4];
    float* out = (float*)d_out;

    const int n_heads = in_sizes[0] / IN_DIM;   // (T+1)
    const int blocks  = (n_heads + WAVES_PER_BLOCK - 1) / WAVES_PER_BLOCK;
    distral_mlp_wmma<<<blocks, WAVES_PER_BLOCK * 32, 0, stream>>>(
        x, W1, b1, W2, b2, out, n_heads);
}